// Phi_r3_82300163326677
// MI455X (gfx1250) — compile-verified
//
#include <hip/hip_runtime.h>

// ---------------------------------------------------------------------------
// Phi_r3 (diff_only) posterior-mean solve for MI455X / gfx1250.
//
//  * Matrix-free Jacobi-PCG: whole 5120-dim state resident in LDS
//    (CDNA5: 320KB/WGP), zero HBM traffic inside the CG loop.
//  * Spatial operator expressed as 32x32 shift-matrix products:
//      A_sym u = c.u - 0.5[H00.(Lu)+L(H00.u)] - 0.5[H11.(uL)+(H11.u)L]
//                - 0.25[H01.(D u D^T) + D(H01.u) D^T]
//    -> every stencil application becomes 8 fp32 32x32x32 matmuls mapped to
//       V_WMMA_F32_16X16X4_F32.
//  * Shift-matrix WMMA operand tiles are loop-invariant (functions of lane
//    ID only): synthesized ONCE into 64 VGPRs before the CG loop and reused
//    by every WMMA (round-1 codegen rebuilt them with ~10 VALU per WMMA).
//  * 1 workgroup per batch item, 5 wave32 waves = 1 wave per time slice.
//  * Deterministic LDS reductions (no FP atomics) -> bitwise reproducible.
// ---------------------------------------------------------------------------

#define NT        5
#define NNODE     1024
#define NTOT      (NT * NNODE)          // 5120
#define NTHREADS  160                   // 5 waves of 32
#define PER_TH    32                    // NTOT / NTHREADS
#define ITERS     350
#define KAPPA2    0.1089f               // 0.33^2
#define INV_R     1000.0f

typedef float v8f __attribute__((ext_vector_type(8)));
typedef float v2f __attribute__((ext_vector_type(2)));

// LDS layout (floats)
#define OFF_C     0
#define OFF_H00   1024
#define OFF_H11   2048
#define OFF_H01   3072
#define OFF_DIAGR 4096                  // 5120
#define OFF_P     9216                  // 5120
#define OFF_Q     14336                 // 5120
#define OFF_Y     19456                 // 5120
#define OFF_T     24576                 // 5 waves x 1024 scratch
#define OFF_R     29696                 // 5120 residual (thread-owned)
#define OFF_D     34816                 // 5120 Jacobi inverse diag
#define OFF_RED   39936                 // reduction buffer
#define SMEM_FLOATS (OFF_RED + 192)     // 40128 floats ~= 157 KB

// L[r][c] = [c==r+1] + [c==r-1] ; D[r][c] = [c==r+1] - [c==r-1]
__device__ __forceinline__ float shift_entry(int r, int c, int diff) {
  float v = (c == r + 1) ? 1.0f : 0.0f;
  v += (c == r - 1) ? (diff ? -1.0f : 1.0f) : 0.0f;
  return v;
}

// OUT = Sh * U (left multiply); tile[h][kc] = Sh[h*16+l15, k] is the WMMA
// A-operand. In-place safe (nb-outer ordering).
__device__ __forceinline__ void wmma_leftmul(const float* U, float* OUT,
                                             const v2f tile[2][8], int lane) {
  const int l15   = lane & 15;
  const int khalf = (lane >> 4) * 2;
  #pragma unroll
  for (int nb = 0; nb < 2; ++nb) {
    const int ncol = nb * 16 + l15;
    v2f B[8];
    #pragma unroll
    for (int kc = 0; kc < 8; ++kc) {
      B[kc].x = U[(kc * 4 + khalf + 0) * 32 + ncol];
      B[kc].y = U[(kc * 4 + khalf + 1) * 32 + ncol];
    }
    #pragma unroll
    for (int mb = 0; mb < 2; ++mb) {
      v8f acc = {0.f, 0.f, 0.f, 0.f, 0.f, 0.f, 0.f, 0.f};
      #pragma unroll
      for (int kc = 0; kc < 8; ++kc)
        acc = __builtin_amdgcn_wmma_f32_16x16x4_f32(false, tile[mb][kc], false,
                                                    B[kc], (short)0, acc,
                                                    false, false);
      const int rbase = mb * 16 + 8 * (lane >> 4);
      #pragma unroll
      for (int v = 0; v < 8; ++v) OUT[(rbase + v) * 32 + ncol] = acc[v];
    }
  }
}

// OUT = U * Sh^T (right multiply); tile[h][kc] = Sh[h*16+l15, k] doubles as
// the WMMA B-operand since B[k][n] = Sh^T[k][n] = Sh[n][k]. In-place safe.
__device__ __forceinline__ void wmma_rightmul(const float* U, float* OUT,
                                              const v2f tile[2][8], int lane) {
  const int l15   = lane & 15;
  const int khalf = (lane >> 4) * 2;
  #pragma unroll
  for (int mb = 0; mb < 2; ++mb) {
    const int mrow = mb * 16 + l15;
    v2f A[8];
    #pragma unroll
    for (int kc = 0; kc < 8; ++kc) {
      A[kc].x = U[mrow * 32 + kc * 4 + khalf + 0];
      A[kc].y = U[mrow * 32 + kc * 4 + khalf + 1];
    }
    #pragma unroll
    for (int nb = 0; nb < 2; ++nb) {
      v8f acc = {0.f, 0.f, 0.f, 0.f, 0.f, 0.f, 0.f, 0.f};
      #pragma unroll
      for (int kc = 0; kc < 8; ++kc)
        acc = __builtin_amdgcn_wmma_f32_16x16x4_f32(false, A[kc], false,
                                                    tile[nb][kc], (short)0, acc,
                                                    false, false);
      const int rbase = mb * 16 + 8 * (lane >> 4);
      const int ncol  = nb * 16 + l15;
      #pragma unroll
      for (int v = 0; v < 8; ++v) OUT[(rbase + v) * 32 + ncol] = acc[v];
    }
  }
}

// out = (I + A_sym) u  (one 32x32 slice, one wave). Force-inlined so the
// precomputed shift tiles stay in VGPRs across the whole CG loop.
__device__ __forceinline__ void m_apply(const float* u, float* out, float* T,
                                        const float* cF, const float* h00,
                                        const float* h11, const float* h01,
                                        const v2f tL[2][8], const v2f tD[2][8],
                                        int lane) {
  for (int p = lane; p < NNODE; p += 32) out[p] = (1.0f + cF[p]) * u[p];
  // H00 (xx) term
  wmma_leftmul(u, T, tL, lane);
  for (int p = lane; p < NNODE; p += 32) out[p] -= 0.5f * h00[p] * T[p];
  for (int p = lane; p < NNODE; p += 32) T[p] = h00[p] * u[p];
  wmma_leftmul(T, T, tL, lane);
  for (int p = lane; p < NNODE; p += 32) out[p] -= 0.5f * T[p];
  // H11 (yy) term
  wmma_rightmul(u, T, tL, lane);
  for (int p = lane; p < NNODE; p += 32) out[p] -= 0.5f * h11[p] * T[p];
  for (int p = lane; p < NNODE; p += 32) T[p] = h11[p] * u[p];
  wmma_rightmul(T, T, tL, lane);
  for (int p = lane; p < NNODE; p += 32) out[p] -= 0.5f * T[p];
  // H01 (xy) cross term
  wmma_leftmul(u, T, tD, lane);
  wmma_rightmul(T, T, tD, lane);
  for (int p = lane; p < NNODE; p += 32) out[p] -= 0.25f * h01[p] * T[p];
  for (int p = lane; p < NNODE; p += 32) T[p] = h01[p] * u[p];
  wmma_leftmul(T, T, tD, lane);
  wmma_rightmul(T, T, tD, lane);
  for (int p = lane; p < NNODE; p += 32) out[p] -= 0.25f * T[p];
}

__device__ float block_reduce(float v, float* red, int tid) {
  __syncthreads();
  red[tid] = v;
  __syncthreads();
  if (tid == 0) {
    float s = 0.0f;
    for (int k = 0; k < NTHREADS; ++k) s += red[k];
    red[NTHREADS] = s;
  }
  __syncthreads();
  return red[NTHREADS];
}

__global__ __launch_bounds__(NTHREADS) void phi_r3_pcg_kernel(
    const float* __restrict__ obs, const int* __restrict__ mask,
    const float* __restrict__ Hf, float* __restrict__ out) {
  extern __shared__ float sm[];
  float* sC   = sm + OFF_C;
  float* sH00 = sm + OFF_H00;
  float* sH11 = sm + OFF_H11;
  float* sH01 = sm + OFF_H01;
  float* sDR  = sm + OFF_DIAGR;
  float* sP   = sm + OFF_P;
  float* sQ   = sm + OFF_Q;
  float* sY   = sm + OFF_Y;
  float* sT   = sm + OFF_T;
  float* sR   = sm + OFF_R;
  float* sD   = sm + OFF_D;
  float* sRed = sm + OFF_RED;

  const int tid  = threadIdx.x;
  const int lane = tid & 31;
  const int wave = tid >> 5;          // == time slice t (5 waves)
  const int b    = blockIdx.x;        // batch item

  __builtin_amdgcn_s_wait_tensorcnt(0);
  __builtin_prefetch(obs + b * NTOT, 0, 0);
  __builtin_prefetch(Hf, 0, 0);

  // ---- loop-invariant WMMA shift tiles (64 VGPRs, computed once) ----
  v2f tL[2][8], tD[2][8];
  {
    const int l15   = lane & 15;
    const int khalf = (lane >> 4) * 2;
    #pragma unroll
    for (int h = 0; h < 2; ++h) {
      const int r = h * 16 + l15;
      #pragma unroll
      for (int kc = 0; kc < 8; ++kc) {
        tL[h][kc].x = shift_entry(r, kc * 4 + khalf + 0, 0);
        tL[h][kc].y = shift_entry(r, kc * 4 + khalf + 1, 0);
        tD[h][kc].x = shift_entry(r, kc * 4 + khalf + 0, 1);
        tD[h][kc].y = shift_entry(r, kc * 4 + khalf + 1, 1);
      }
    }
  }

  // ---- stationary fields (shared by both batches) ----
  for (int p = tid; p < NNODE; p += NTHREADS) {
    const float a00 = Hf[p];              // H[0][0]
    const float a01 = Hf[NNODE + p];      // H[0][1]
    const float a11 = Hf[3 * NNODE + p];  // H[1][1]
    sH00[p] = a00;
    sH01[p] = a01;
    sH11[p] = a11;
    sC[p]   = KAPPA2 + 2.0f * a00 + 2.0f * a11;
  }

  // ---- rhs + observation diag (thread-owned striding) ----
  float xloc[PER_TH];
  #pragma unroll
  for (int j = 0; j < PER_TH; ++j) {
    const int i = tid + NTHREADS * j;
    const int t = i >> 10, p = i & (NNODE - 1);
    const int xg = p >> 5, yg = p & 31;
    const int gi = ((b * NT + t) * 32 + yg) * 32 + xg;   // input [b][t][y][x]
    const float mf = (mask[gi] != 0) ? 1.0f : 0.0f;
    const float dr = INV_R * mf;
    sDR[i]  = dr;
    sR[i]   = dr * obs[gi];   // r0 = b (x0 = 0)
    xloc[j] = 0.0f;
  }
  __syncthreads();

  // ---- exact Jacobi preconditioner: diag(Qxx) = sum_q Msym[p,q]^2 (+I,+R) ----
  float rz = 0.0f;
  #pragma unroll
  for (int j = 0; j < PER_TH; ++j) {
    const int i = tid + NTHREADS * j;
    const int t = i >> 10, p = i & (NNODE - 1);
    const int xg = p >> 5, yg = p & 31;
    const float m00 = 1.0f + sC[p];
    float s = m00 * m00;
    if (xg < 31) { float e = 0.5f  * (sH00[p] + sH00[p + 32]); s += e * e; }
    if (xg > 0)  { float e = 0.5f  * (sH00[p] + sH00[p - 32]); s += e * e; }
    if (yg < 31) { float e = 0.5f  * (sH11[p] + sH11[p + 1]);  s += e * e; }
    if (yg > 0)  { float e = 0.5f  * (sH11[p] + sH11[p - 1]);  s += e * e; }
    if (xg < 31 && yg < 31) { float e = 0.25f * (sH01[p] + sH01[p + 33]); s += e * e; }
    if (xg > 0  && yg > 0)  { float e = 0.25f * (sH01[p] + sH01[p - 33]); s += e * e; }
    if (xg < 31 && yg > 0)  { float e = 0.25f * (sH01[p] + sH01[p + 31]); s += e * e; }
    if (xg > 0  && yg < 31) { float e = 0.25f * (sH01[p] + sH01[p - 31]); s += e * e; }
    if (t >= 1 && t <= NT - 2) s += 1.0f;                 // MM+I middle blocks
    s += sDR[i];
    const float d = 1.0f / s;
    const float r0 = sR[i];
    sD[i] = d;
    sP[i] = d * r0;                                       // p0 = Minv r0
    rz += r0 * d * r0;
  }
  rz = block_reduce(rz, sRed, tid);

  // ---- PCG loop ----
  for (int it = 0; it < ITERS; ++it) {
    __syncthreads();                                      // sP stable
    // phase 1: y_t = M p_t (one wave per slice, all WMMA)
    m_apply(sP + wave * NNODE, sY + wave * NNODE, sT + wave * NNODE,
            sC, sH00, sH11, sH01, tL, tD, lane);
    __syncthreads();
    // phase 2: q_t = M y_t + temporal coupling + obs diag
    m_apply(sY + wave * NNODE, sQ + wave * NNODE, sT + wave * NNODE,
            sC, sH00, sH11, sH01, tL, tD, lane);
    for (int p = lane; p < NNODE; p += 32) {
      const int i = wave * NNODE + p;
      float a = sQ[i];
      if (wave >= 1 && wave <= NT - 2) a += sP[i];        // +I on middle blocks
      if (wave > 0)      a -= sY[i - NNODE];              // -M p_{t-1}
      if (wave < NT - 1) a -= sY[i + NNODE];              // -M p_{t+1}
      a += sDR[i] * sP[i];                                // + R^-1 mask
      sQ[i] = a;
    }
    __syncthreads();                                      // sQ visible
    float pq = 0.0f;
    #pragma unroll
    for (int j = 0; j < PER_TH; ++j) {
      const int i = tid + NTHREADS * j;
      pq += sP[i] * sQ[i];
    }
    pq = block_reduce(pq, sRed, tid);
    const float alpha = (pq != 0.0f) ? (rz / pq) : 0.0f;
    float rznew = 0.0f;
    #pragma unroll
    for (int j = 0; j < PER_TH; ++j) {
      const int i = tid + NTHREADS * j;
      xloc[j] += alpha * sP[i];
      const float r = sR[i] - alpha * sQ[i];
      sR[i] = r;
      rznew += r * sD[i] * r;
    }
    rznew = block_reduce(rznew, sRed, tid);
    const float beta = (rz != 0.0f) ? (rznew / rz) : 0.0f;
    #pragma unroll
    for (int j = 0; j < PER_TH; ++j) {
      const int i = tid + NTHREADS * j;
      sP[i] = sD[i] * sR[i] + beta * sP[i];
    }
    rz = rznew;
  }

  // ---- outputs ----
  // x_out[b][t][y][x] = x[t*1024 + x*32 + y]
  #pragma unroll
  for (int j = 0; j < PER_TH; ++j) {
    const int i = tid + NTHREADS * j;
    const int t = i >> 10, p = i & (NNODE - 1);
    const int xg = p >> 5, yg = p & 31;
    out[b * NTOT + t * NNODE + yg * 32 + xg] = xloc[j];
  }
  // H_out[b][i][j][x][y] = H_field[i][j][x*32+y] (broadcast over b)
  for (int k = tid; k < 4 * NNODE; k += NTHREADS)
    out[2 * NTOT + b * 4 * NNODE + k] = Hf[k];
}

extern "C" void kernel_launch(void* const* d_in, const int* in_sizes, int n_in,
                              void* d_out, int out_size, void* d_ws, size_t ws_size,
                              hipStream_t stream) {
  (void)in_sizes; (void)n_in; (void)d_ws; (void)ws_size; (void)out_size;
  const float* obs  = (const float*)d_in[1];   // x (d_in[0]) is unused by the math
  const int*   mask = (const int*)d_in[2];
  const float* Hf   = (const float*)d_in[3];
  float* out = (float*)d_out;
  phi_r3_pcg_kernel<<<dim3(2), dim3(NTHREADS), SMEM_FLOATS * sizeof(float), stream>>>(
      obs, mask, Hf, out);
}